// GNNStack_32839319945861
// MI455X (gfx1250) — compile-verified
//
#include <hip/hip_runtime.h>
#include <hip/hip_bf16.h>
#include <math.h>

// GATStack on MI455X (gfx1250, wave32).
// Bottleneck analysis: edge scatter/gather (~6 GB irregular traffic) dominates;
// dense GEMMs (~12 GFLOP) are negligible at WMMA rates -> use full-precision
// f32 WMMA (V_WMMA_F32_16X16X4_F32) for the GEMMs, and 2 edge passes per layer.

#define HC 128          // H*C hidden width (== IN)
#define CC 64           // projection hidden dim
#define NOUT 40         // classes
#define EPSV 1e-16f

typedef float    v2f  __attribute__((ext_vector_type(2)));
typedef float    v8f  __attribute__((ext_vector_type(8)));

// ---------------------------------------------------------------------------
// GEMM: C[M,Nout] = A[M,K] * W[Nout,K]^T + bias   (W row-major [Nout,K])
// One wave (32 lanes) per 16x16 output tile. M % 16 == 0 (100000/16 = 6250).
// V_WMMA_F32_16X16X4_F32, K stepped by 4:
//   A 16x4 : lanes 0-15 -> M=lane, VGPR{0,1}=K{0,1}; lanes 16-31 -> K{2,3}
//   B 4x16 : lanes 0-15 -> N=lane, VGPR{0,1}=K{0,1}; lanes 16-31 -> K{2,3}
//   C 16x16: VGPR r -> M = r (+8 for upper half-wave), N = lane&15
// The tile-edge case (only the last tile of the 40-class GEMM) is handled with
// a uniform per-block branch + branch-free clamp/mask loads so the inner loop
// never touches EXEC (WMMA requires EXEC all-ones anyway).
// ---------------------------------------------------------------------------
__global__ void gemm_wmma(const float* __restrict__ A, const float* __restrict__ W,
                          const float* __restrict__ bias, float* __restrict__ Cm,
                          int M, int K, int Nout, int relu)
{
  const int lane  = threadIdx.x & 31;
  const int half  = lane >> 4;
  const int l15   = lane & 15;
  const int mBase = blockIdx.x * 16;
  const int nBase = blockIdx.y * 16;
  const int col   = nBase + l15;         // output column this lane owns (B/C/D)
  v8f acc = {};

  const float* ap = A + (size_t)(mBase + l15) * K + 2 * half;

  if (nBase + 16 <= Nout) {
    // Full tile: clean, branch-free, software-pipelined loop.
    const float* wp = W + (size_t)col * K + 2 * half;
#pragma unroll 8
    for (int k = 0; k < K; k += 4) {
      v2f a; a.x = ap[0]; a.y = ap[1];
      v2f b; b.x = wp[0]; b.y = wp[1];
      ap += 4; wp += 4;
      acc = __builtin_amdgcn_wmma_f32_16x16x4_f32(false, a, false, b,
                                                  (short)0, acc, false, false);
    }
  } else {
    // Partial tile: clamp to a valid W row, zero the fragment via multiply.
    const int   colC = (col < Nout) ? col : (Nout - 1);
    const float msk  = (col < Nout) ? 1.f : 0.f;
    const float* wp  = W + (size_t)colC * K + 2 * half;
#pragma unroll 8
    for (int k = 0; k < K; k += 4) {
      v2f a; a.x = ap[0];       a.y = ap[1];
      v2f b; b.x = wp[0] * msk; b.y = wp[1] * msk;
      ap += 4; wp += 4;
      acc = __builtin_amdgcn_wmma_f32_16x16x4_f32(false, a, false, b,
                                                  (short)0, acc, false, false);
    }
  }

  if (col < Nout) {
    const float bv = bias[col];
#pragma unroll
    for (int r = 0; r < 8; ++r) {
      const int mrow = mBase + r + 8 * half;
      float v = acc[r] + bv;
      if (relu) v = fmaxf(v, 0.f);
      Cm[(size_t)mrow * Nout + col] = v;
    }
  }
}

// ---------------------------------------------------------------------------
// Order-preserving float <-> uint mapping for atomicMax-based segment max.
// ---------------------------------------------------------------------------
__device__ __forceinline__ unsigned f2ord(float f) {
  unsigned u = __float_as_uint(f);
  return (u & 0x80000000u) ? ~u : (u | 0x80000000u);
}
__device__ __forceinline__ float ord2f(unsigned o) {
  unsigned u = (o & 0x80000000u) ? (o & 0x7fffffffu) : ~o;
  return __uint_as_float(u);
}

// Pass 1: per-(dst,channel) segment max of e = ar*h[dst] + al*h[src].
// One thread = one edge x 4 channels (float4 traffic, 512B/wave coalesced on h).
__global__ void edge_max_kernel(const float* __restrict__ h,
                                const int* __restrict__ src, const int* __restrict__ dst,
                                const float* __restrict__ al, const float* __restrict__ ar,
                                unsigned* __restrict__ m, int E)
{
  const long long gid = (long long)blockIdx.x * blockDim.x + threadIdx.x;
  const int e = (int)(gid >> 5);
  if (e >= E) return;
  const int c4 = ((int)gid & 31) * 4;
  const int s = src[e], d = dst[e];
  const float4 hs = *(const float4*)(h + (size_t)s * HC + c4);
  const float4 hd = *(const float4*)(h + (size_t)d * HC + c4);
  const float4 av = *(const float4*)(al + c4);
  const float4 rv = *(const float4*)(ar + c4);
  unsigned* mp = m + (size_t)d * HC + c4;
  atomicMax(mp + 0, f2ord(rv.x * hd.x + av.x * hs.x));
  atomicMax(mp + 1, f2ord(rv.y * hd.y + av.y * hs.y));
  atomicMax(mp + 2, f2ord(rv.z * hd.z + av.z * hs.z));
  atomicMax(mp + 3, f2ord(rv.w * hd.w + av.w * hs.w));
}

// Pass 2 (fused): w = exp(e - m[dst]); s[dst] += w; num[dst] += w * h[src].
// out = num / (s + eps) is separable since s is constant per segment+channel.
__global__ void edge_sum_kernel(const float* __restrict__ h,
                                const int* __restrict__ src, const int* __restrict__ dst,
                                const float* __restrict__ al, const float* __restrict__ ar,
                                const unsigned* __restrict__ m,
                                float* __restrict__ sbuf, float* __restrict__ numbuf, int E)
{
  const long long gid = (long long)blockIdx.x * blockDim.x + threadIdx.x;
  const int e = (int)(gid >> 5);
  if (e >= E) return;
  const int c4 = ((int)gid & 31) * 4;
  const int s = src[e], d = dst[e];
  const float4 hs = *(const float4*)(h + (size_t)s * HC + c4);
  const float4 hd = *(const float4*)(h + (size_t)d * HC + c4);
  const float4 av = *(const float4*)(al + c4);
  const float4 rv = *(const float4*)(ar + c4);
  const unsigned* mp = m + (size_t)d * HC + c4;
  const float w0 = __expf(rv.x * hd.x + av.x * hs.x - ord2f(mp[0]));
  const float w1 = __expf(rv.y * hd.y + av.y * hs.y - ord2f(mp[1]));
  const float w2 = __expf(rv.z * hd.z + av.z * hs.z - ord2f(mp[2]));
  const float w3 = __expf(rv.w * hd.w + av.w * hs.w - ord2f(mp[3]));
  float* sp = sbuf + (size_t)d * HC + c4;
  float* np = numbuf + (size_t)d * HC + c4;
  atomicAdd(sp + 0, w0); atomicAdd(np + 0, w0 * hs.x);
  atomicAdd(sp + 1, w1); atomicAdd(np + 1, w1 * hs.y);
  atomicAdd(sp + 2, w2); atomicAdd(np + 2, w2 * hs.z);
  atomicAdd(sp + 3, w3); atomicAdd(np + 3, w3 * hs.w);
}

// Pass 3: h_out = relu(num / (s + eps)) per node/channel.
__global__ void finalize_kernel(const float* __restrict__ numbuf,
                                const float* __restrict__ sbuf,
                                float* __restrict__ hout, int Nn)
{
  const long long gid = (long long)blockIdx.x * blockDim.x + threadIdx.x;
  const int n = (int)(gid >> 5);
  if (n >= Nn) return;
  const int c4 = ((int)gid & 31) * 4;
  const float4 nu = *(const float4*)(numbuf + (size_t)n * HC + c4);
  const float4 sv = *(const float4*)(sbuf + (size_t)n * HC + c4);
  float4 o;
  o.x = fmaxf(nu.x / (sv.x + EPSV), 0.f);
  o.y = fmaxf(nu.y / (sv.y + EPSV), 0.f);
  o.z = fmaxf(nu.z / (sv.z + EPSV), 0.f);
  o.w = fmaxf(nu.w / (sv.w + EPSV), 0.f);
  *(float4*)(hout + (size_t)n * HC + c4) = o;
}

// In-place log_softmax over rows of NOUT=40: one wave per row.
__global__ void log_softmax_kernel(float* __restrict__ logits, int Nn)
{
  const int wid  = (int)(((long long)blockIdx.x * blockDim.x + threadIdx.x) >> 5);
  const int lane = threadIdx.x & 31;
  if (wid >= Nn) return;
  float* row = logits + (size_t)wid * NOUT;
  const float x0 = (lane < NOUT)      ? row[lane]      : -INFINITY;
  const float x1 = (lane + 32 < NOUT) ? row[lane + 32] : -INFINITY;
  float mx = fmaxf(x0, x1);
#pragma unroll
  for (int off = 16; off > 0; off >>= 1) mx = fmaxf(mx, __shfl_xor(mx, off, 32));
  float s = ((lane < NOUT) ? __expf(x0 - mx) : 0.f)
          + ((lane + 32 < NOUT) ? __expf(x1 - mx) : 0.f);
#pragma unroll
  for (int off = 16; off > 0; off >>= 1) s += __shfl_xor(s, off, 32);
  const float lse = __logf(s) + mx;
  if (lane < NOUT)      row[lane]      = x0 - lse;
  if (lane + 32 < NOUT) row[lane + 32] = x1 - lse;
}

extern "C" void kernel_launch(void* const* d_in, const int* in_sizes, int n_in,
                              void* d_out, int out_size, void* d_ws, size_t ws_size,
                              hipStream_t stream) {
  const float* x   = (const float*)d_in[0];
  const int*   ei  = (const int*)d_in[1];
  const int Nn = in_sizes[0] / HC;    // 100000
  const int E  = in_sizes[1] / 2;     // 800000
  const int* srcIdx = ei;
  const int* dstIdx = ei + E;

  const float* Wl[3]  = {(const float*)d_in[2],  (const float*)d_in[6],  (const float*)d_in[10]};
  const float* bl[3]  = {(const float*)d_in[3],  (const float*)d_in[7],  (const float*)d_in[11]};
  const float* alv[3] = {(const float*)d_in[4],  (const float*)d_in[8],  (const float*)d_in[12]};
  const float* arv[3] = {(const float*)d_in[5],  (const float*)d_in[9],  (const float*)d_in[13]};
  const float* Wp1 = (const float*)d_in[14];
  const float* bp1 = (const float*)d_in[15];
  const float* Wp2 = (const float*)d_in[16];
  const float* bp2 = (const float*)d_in[17];

  // Workspace: hA, hB, m, s, num (N*128 f32 each) + p1 (N*64 f32).
  const size_t NB = (size_t)Nn * HC * sizeof(float);
  char* w = (char*)d_ws;
  float*    hA     = (float*)(w);
  float*    hB     = (float*)(w + NB);
  unsigned* mbuf   = (unsigned*)(w + 2 * NB);
  float*    sbuf   = (float*)(w + 3 * NB);
  float*    numbuf = (float*)(w + 4 * NB);
  float*    p1     = (float*)(w + 5 * NB);

  const int edgeBlocks = (int)(((long long)E * 32 + 255) / 256);
  const int nodeBlocks = (int)(((long long)Nn * 32 + 255) / 256);

  const float* cur = x;
  for (int l = 0; l < 3; ++l) {
    hipMemsetAsync(mbuf,   0, NB, stream);   // 0 == ord-encoded minimum
    hipMemsetAsync(sbuf,   0, NB, stream);
    hipMemsetAsync(numbuf, 0, NB, stream);
    gemm_wmma<<<dim3(Nn / 16, HC / 16), 32, 0, stream>>>(cur, Wl[l], bl[l], hB,
                                                         Nn, HC, HC, 0);
    edge_max_kernel<<<edgeBlocks, 256, 0, stream>>>(hB, srcIdx, dstIdx,
                                                    alv[l], arv[l], mbuf, E);
    edge_sum_kernel<<<edgeBlocks, 256, 0, stream>>>(hB, srcIdx, dstIdx,
                                                    alv[l], arv[l], mbuf,
                                                    sbuf, numbuf, E);
    finalize_kernel<<<nodeBlocks, 256, 0, stream>>>(numbuf, sbuf, hA, Nn);
    cur = hA;
  }
  // Post-MLP: 128 -> 64 -> 40, then row log_softmax (in-place on d_out).
  gemm_wmma<<<dim3(Nn / 16, CC / 16), 32, 0, stream>>>(hA, Wp1, bp1, p1,
                                                       Nn, HC, CC, 0);
  gemm_wmma<<<dim3(Nn / 16, (NOUT + 15) / 16), 32, 0, stream>>>(p1, Wp2, bp2,
                                                                (float*)d_out,
                                                                Nn, CC, NOUT, 0);
  log_softmax_kernel<<<(int)(((long long)Nn * 32 + 255) / 256), 256, 0, stream>>>(
      (float*)d_out, Nn);
}